// ModifiedMarginRankingLoss_47425028882481
// MI455X (gfx1250) — compile-verified
//
#include <hip/hip_runtime.h>

// ModifiedMarginRankingLoss, MI455X (gfx1250, wave32).
//
// Closed-form reduction: with u_i = s_i - MARGIN*i (i = label-sorted pos),
//   sum_{i<j} relu(u_i - u_j) = sum_k u_k * (valrank_k - labelrank_k)
// so per row we only need two stable ranks (unique u64 keys -> one
// v_cmp_lt_u64 per comparison, LDS reads are wave-uniform broadcasts).
// CDNA5 path: async global->LDS staging of the label row (ASYNCcnt).

#define LROW 128
#define MARGIN 0.1f

#if defined(__has_builtin)
#  if __has_builtin(__builtin_amdgcn_global_load_async_to_lds_b32) && \
      __has_builtin(__builtin_amdgcn_s_wait_asynccnt)
#    define HAS_ASYNC_LDS 1
#  endif
#endif

#define GAS __attribute__((address_space(1)))
#define LAS __attribute__((address_space(3)))

__device__ __forceinline__ unsigned sortable_bits(float f) {
  unsigned u = __float_as_uint(f);
  // monotonic float -> unsigned map (negatives: flip all; positives: flip sign)
  return u ^ ((unsigned)((int)u >> 31) | 0x80000000u);
}

__global__ void __launch_bounds__(LROW)
row_loss_kernel(const float* __restrict__ scores,
                const float* __restrict__ labels,
                float* __restrict__ partial,
                float inv_batch) {
  __shared__ float raw_lab[LROW];               // async staging target
  __shared__ unsigned long long keys[LROW];     // (sortable_key<<32)|idx
  __shared__ float wave_sum[LROW / 32];

  const int tid = threadIdx.x;
  const long long base = (long long)blockIdx.x * LROW;

  // Stage this row's labels into LDS via CDNA5 async copy (ASYNCcnt path),
  // overlapping with the register load of the score.
#if defined(HAS_ASYNC_LDS)
  __builtin_amdgcn_global_load_async_to_lds_b32(
      (GAS int*)(labels + base + tid), (LAS int*)&raw_lab[tid], 0, 0);
#else
  raw_lab[tid] = labels[base + tid];
#endif
  const float sc = scores[base + tid];
#if defined(HAS_ASYNC_LDS)
  __builtin_amdgcn_s_wait_asynccnt(0);
#endif
  __syncthreads();

  // Pass 1: stable rank of this element's label (ascending, tie-break by idx).
  {
    unsigned kb = sortable_bits(raw_lab[tid]);
    keys[tid] = ((unsigned long long)kb << 32) | (unsigned)tid;
  }
  __syncthreads();

  const unsigned long long mykey = keys[tid];
  int r = 0;
#pragma unroll 8
  for (int m = 0; m < LROW; ++m) r += (keys[m] < mykey) ? 1 : 0;

  // u = score - MARGIN * labelrank
  const float u = fmaf(-MARGIN, (float)r, sc);

  __syncthreads();  // done reading label keys; reuse the array

  // Pass 2: stable rank of u among the row's u values.
  {
    unsigned ub = sortable_bits(u);
    keys[tid] = ((unsigned long long)ub << 32) | (unsigned)tid;
  }
  __syncthreads();

  const unsigned long long ukey = keys[tid];
  int q = 0;
#pragma unroll 8
  for (int m = 0; m < LROW; ++m) q += (keys[m] < ukey) ? 1 : 0;

  // Per-element contribution to sum_{i<j} relu(...)
  float contrib = u * (float)(q - r);

  // wave32 reduction, then combine the 4 waves.
#pragma unroll
  for (int off = 16; off > 0; off >>= 1)
    contrib += __shfl_down(contrib, off, 32);
  if ((tid & 31) == 0) wave_sum[tid >> 5] = contrib;
  __syncthreads();
  if (tid == 0) {
    float s = wave_sum[0] + wave_sum[1] + wave_sum[2] + wave_sum[3];
    partial[blockIdx.x] = s * inv_batch;   // fold mean over batch here
  }
}

__global__ void __launch_bounds__(256)
final_reduce_kernel(const float* __restrict__ partial, int n,
                    float* __restrict__ out) {
  __shared__ float buf[256];
  const int tid = threadIdx.x;
  float s = 0.0f;
  for (int i = tid; i < n; i += 256) s += partial[i];
  buf[tid] = s;
  __syncthreads();
#pragma unroll
  for (int step = 128; step > 0; step >>= 1) {
    if (tid < step) buf[tid] += buf[tid + step];
    __syncthreads();
  }
  if (tid == 0) out[0] = buf[0];   // single deterministic writer; no pre-zero needed
}

extern "C" void kernel_launch(void* const* d_in, const int* in_sizes, int n_in,
                              void* d_out, int out_size, void* d_ws, size_t ws_size,
                              hipStream_t stream) {
  const float* scores = (const float*)d_in[0];
  const float* labels = (const float*)d_in[1];
  const int rows = in_sizes[0] / LROW;          // B = 8192
  float* partial = (float*)d_ws;                // rows floats of scratch
  float* out = (float*)d_out;

  row_loss_kernel<<<rows, LROW, 0, stream>>>(scores, labels, partial,
                                             1.0f / (float)rows);
  final_reduce_kernel<<<1, 256, 0, stream>>>(partial, rows, out);
}